// BPCAPooling_25769803830
// MI455X (gfx1250) — compile-verified
//
#include <hip/hip_runtime.h>
#include <hip/hip_bf16.h>
#include <math.h>

// BPCA pooling on MI455X (gfx1250).
//
// Per batch b (input slice 64*64*512 f32 = 8 MB):
//   X = slice viewed as (524288 x 4) rows of 4 consecutive floats.
//   mean = col-mean(X); cov = X^T X / n - mean mean^T (4x4)
//   top  = eigvec of largest eigenvalue of cov
//   out[b,h1,w1,(h2*2+w2)*128+k] = dot(in[b,2h1+h2,2w1+w2,4k:4k+4] - mean, top)
//
// Bandwidth-bound: 256 MB read x2 + 64 MB write ~ 25 us floor @ 23.3 TB/s.
// Gram matrix X^T X accumulated with V_WMMA_F32_16X16X4_F32: view the stream
// as Y (n/4 x 16); D += A(16x4) x B(4x16) with A = chunk^T, B = chunk. The
// wave32 A layout (lane L: v0=K0/K2, v1=K1/K3) and B layout (rows striped
// across lanes) are transpose mirrors, so A and B share the SAME two VGPRs.
// The four 4x4 diagonal blocks of Y^T Y sum to X^T X. Inner loop unrolled 8x
// so 16 independent b32 loads can be claused ahead of the 8 WMMAs (D->C
// accumulation chains are hazard-free; only D->A/B needs NOPs).

typedef __attribute__((ext_vector_type(2))) float v2f;
typedef __attribute__((ext_vector_type(8))) float v8f;

#define ELEMS_PER_BATCH (64 * 64 * 512)          // 2097152
#define CHUNKS_PER_BATCH (ELEMS_PER_BATCH / 64)  // 32768 chunks of 64 floats
#define BLOCKS_PER_BATCH 8
#define WAVES_PER_BLOCK 8
#define WAVES_PER_BATCH (BLOCKS_PER_BATCH * WAVES_PER_BLOCK)  // 64
#define UNROLL 8
#define GROUPS_PER_BATCH (CHUNKS_PER_BATCH / UNROLL)          // 4096
#define ITERS (GROUPS_PER_BATCH / WAVES_PER_BATCH)            // 64
#define N_ROWS 524288.0f
// ws layout (floats):
//  [ (b*8+blk)*20 .. +20 )  : per-block partials: sum[4], gram16[16]
//  [ 5120 + b*8 .. +8 )     : finalized: top[4], bias, pad
#define WS_FINAL_OFF 5120

__global__ void bpca_stats(const float* __restrict__ in, float* __restrict__ ws) {
  const int b    = blockIdx.y;
  const int lane = threadIdx.x & 31;
  const int w    = threadIdx.x >> 5;                 // wave in block, 0..7
  const int wid  = blockIdx.x * WAVES_PER_BLOCK + w; // wave in batch, 0..63
  const size_t base = (size_t)b * ELEMS_PER_BATCH;
  // WMMA f32 16x16x4 A-matrix layout: lane L<16 holds A[L][0],A[L][1] in
  // v0,v1; lane L+16 holds A[L][2],A[L][3].  With A[m][k]=in[c*64+16k+m]:
  const int laneOff = (lane & 15) + ((lane >> 4) << 5); // +32 for hi half

  v8f acc = {};        // 16x16 f32 accumulator (Y^T Y partial)
  float sacc = 0.0f;   // per-lane running sum (this lane touches col lane%4)

  // Uniform trip count: every wave does exactly ITERS iterations of
  // UNROLL chunks (2 KB contiguous per wave-iteration).
  for (int it = 0; it < ITERS; ++it) {
    const int g = wid + it * WAVES_PER_BATCH;            // group of 8 chunks
    const float* p = in + base + (size_t)g * (64 * UNROLL) + laneOff;
    // Prefetch this wave's next group (lane addresses fall only in the group's first 64-float chunk).
    __builtin_prefetch(p + (size_t)WAVES_PER_BATCH * 64 * UNROLL, 0, 1);
    float a0[UNROLL], a1[UNROLL];
#pragma unroll
    for (int u = 0; u < UNROLL; ++u) {
      a0[u] = p[u * 64];
      a1[u] = p[u * 64 + 16];
    }
#pragma unroll
    for (int u = 0; u < UNROLL; ++u) {
      v2f ab; ab.x = a0[u]; ab.y = a1[u];
      acc = __builtin_amdgcn_wmma_f32_16x16x4_f32(
          /*neg_a=*/false, ab, /*neg_b=*/false, ab,
          /*c_mod=*/(short)0, acc, /*reuse_a=*/false, /*reuse_b=*/false);
      sacc += a0[u] + a1[u];
    }
  }

  // Deterministic block reduction through LDS (no atomics).
  __shared__ float ldsD[WAVES_PER_BLOCK * 256]; // all D elements per wave
  __shared__ float ldsS[WAVES_PER_BLOCK * 32];  // per-lane sums
#pragma unroll
  for (int v = 0; v < 8; ++v) ldsD[w * 256 + v * 32 + lane] = acc[v];
  ldsS[w * 32 + lane] = sacc;
  __syncthreads();

  const int t = threadIdx.x;
  float* wsPart = ws + (size_t)(b * BLOCKS_PER_BATCH + blockIdx.x) * 20;
  if (t < 16) {
    // gram[i][j] = sum over diagonal 4x4 blocks of D.
    // D[M][N] lives in vgpr (M&7), lane (N + 16*(M>>3)).
    const int i = t >> 2, j = t & 3;
    float tot = 0.0f;
    for (int wv = 0; wv < WAVES_PER_BLOCK; ++wv) {
#pragma unroll
      for (int blk = 0; blk < 4; ++blk) {
        const int M = blk * 4 + i;
        const int N = blk * 4 + j;
        tot += ldsD[wv * 256 + (M & 7) * 32 + (N + ((M >> 3) << 4))];
      }
    }
    wsPart[4 + t] = tot;
  } else if (t < 20) {
    const int j = t - 16;  // column index = lane % 4
    float tot = 0.0f;
    for (int wv = 0; wv < WAVES_PER_BLOCK; ++wv)
      for (int ln = j; ln < 32; ln += 4) tot += ldsS[wv * 32 + ln];
    wsPart[j] = tot;
  }
}

// One block per batch: merge partials (fixed order), 4x4 Jacobi eigensolve.
__global__ void bpca_finalize(float* __restrict__ ws) {
  const int b = blockIdx.x;
  if (threadIdx.x != 0) return;

  float s4[4] = {0.f, 0.f, 0.f, 0.f};
  float g[16];
  for (int k = 0; k < 16; ++k) g[k] = 0.f;
  for (int blk = 0; blk < BLOCKS_PER_BATCH; ++blk) {
    const float* p = ws + (size_t)(b * BLOCKS_PER_BATCH + blk) * 20;
    for (int j = 0; j < 4; ++j) s4[j] += p[j];
    for (int k = 0; k < 16; ++k) g[k] += p[4 + k];
  }

  const float invn = 1.0f / N_ROWS;
  float mean[4];
  for (int j = 0; j < 4; ++j) mean[j] = s4[j] * invn;

  float a[4][4], vM[4][4];
  for (int i = 0; i < 4; ++i)
    for (int j = 0; j < 4; ++j) {
      a[i][j]  = g[i * 4 + j] * invn - mean[i] * mean[j];
      vM[i][j] = (i == j) ? 1.0f : 0.0f;
    }

  // Cyclic Jacobi for symmetric 4x4.
  for (int sweep = 0; sweep < 16; ++sweep) {
    for (int p = 0; p < 3; ++p) {
      for (int q = p + 1; q < 4; ++q) {
        const float apq = a[p][q];
        if (fabsf(apq) < 1e-12f) continue;
        const float theta = (a[q][q] - a[p][p]) / (2.0f * apq);
        const float tt = (theta >= 0.0f ? 1.0f : -1.0f) /
                         (fabsf(theta) + sqrtf(theta * theta + 1.0f));
        const float cc = 1.0f / sqrtf(tt * tt + 1.0f);
        const float ss = tt * cc;
        for (int k = 0; k < 4; ++k) {
          const float akp = a[k][p], akq = a[k][q];
          a[k][p] = cc * akp - ss * akq;
          a[k][q] = ss * akp + cc * akq;
        }
        for (int k = 0; k < 4; ++k) {
          const float apk = a[p][k], aqk = a[q][k];
          a[p][k] = cc * apk - ss * aqk;
          a[q][k] = ss * apk + cc * aqk;
        }
        for (int k = 0; k < 4; ++k) {
          const float vkp = vM[k][p], vkq = vM[k][q];
          vM[k][p] = cc * vkp - ss * vkq;
          vM[k][q] = ss * vkp + cc * vkq;
        }
      }
    }
  }

  int qb = 0;
  float best = a[0][0];
  for (int q = 1; q < 4; ++q)
    if (a[q][q] > best) { best = a[q][q]; qb = q; }
  float top[4];
  for (int i = 0; i < 4; ++i) top[i] = vM[i][qb];
  // Sign convention: largest-|component| positive.
  int im = 0;
  float am = fabsf(top[0]);
  for (int i = 1; i < 4; ++i) {
    const float af = fabsf(top[i]);
    if (af > am) { am = af; im = i; }
  }
  if (top[im] < 0.0f)
    for (int i = 0; i < 4; ++i) top[i] = -top[i];

  const float bias = mean[0] * top[0] + mean[1] * top[1] +
                     mean[2] * top[2] + mean[3] * top[3];
  float* o = ws + WS_FINAL_OFF + (size_t)b * 8;
  for (int i = 0; i < 4; ++i) o[i] = top[i];
  o[4] = bias;
}

// grid (512, 32), 256 threads: 4 consecutive outputs per thread (same pixel,
// k..k+3): 4x global_load_b128 (64 B/lane contiguous) + 1x global_store_b128.
__global__ void bpca_transform(const float* __restrict__ in,
                               const float* __restrict__ ws,
                               float* __restrict__ out) {
  const int b = blockIdx.y;
  const float* f = ws + WS_FINAL_OFF + (size_t)b * 8;
  const float t0 = f[0], t1 = f[1], t2 = f[2], t3 = f[3], bias = f[4];

  const int o4 = blockIdx.x * 256 + (int)threadIdx.x; // float4 group, 0..131071
  const int o  = o4 * 4;                              // 0..524287 within batch
  const int co = o & 511;          // multiple of 4
  const int w1 = (o >> 9) & 31;
  const int h1 = o >> 14;
  const int k  = co & 127;         // multiple of 4; k..k+3 stay in this pixel
  const int w2 = (co >> 7) & 1;
  const int h2 = co >> 8;

  const size_t inIdx =
      ((((size_t)b * 64 + 2 * h1 + h2) * 64) + 2 * w1 + w2) * 512 + 4 * k;
  const float4* pv = (const float4*)(in + inIdx);   // 16 consecutive floats
  const float4 v0 = pv[0], v1 = pv[1], v2 = pv[2], v3 = pv[3];

  float4 r;
  r.x = v0.x * t0 + v0.y * t1 + v0.z * t2 + v0.w * t3 - bias;
  r.y = v1.x * t0 + v1.y * t1 + v1.z * t2 + v1.w * t3 - bias;
  r.z = v2.x * t0 + v2.y * t1 + v2.z * t2 + v2.w * t3 - bias;
  r.w = v3.x * t0 + v3.y * t1 + v3.z * t2 + v3.w * t3 - bias;
  *(float4*)(out + (size_t)b * 524288 + o) = r;
}

extern "C" void kernel_launch(void* const* d_in, const int* in_sizes, int n_in,
                              void* d_out, int out_size, void* d_ws, size_t ws_size,
                              hipStream_t stream) {
  (void)in_sizes; (void)n_in; (void)out_size; (void)ws_size;
  const float* in = (const float*)d_in[0];
  float* out = (float*)d_out;
  float* ws  = (float*)d_ws;   // needs 5120 + 32*8 = 5376 floats (~21 KB)

  bpca_stats    <<<dim3(BLOCKS_PER_BATCH, 32), 256, 0, stream>>>(in, ws);
  bpca_finalize <<<32, 32, 0, stream>>>(ws);
  bpca_transform<<<dim3(512, 32), 256, 0, stream>>>(in, ws, out);
}